// Net_14199161880903
// MI455X (gfx1250) — compile-verified
//
#include <hip/hip_runtime.h>

typedef float v2f __attribute__((ext_vector_type(2)));
typedef float v8f __attribute__((ext_vector_type(8)));

#define NNODES 100000
#define NEDGES 1000000
#define INF    64
#define HIDF   128
#define NCLS   40

// ---------------- workspace layout (fp32 words) -----------------------------
// agg1 : [0,           6,400,000)   100000*64
// agg2 : [6,400,000,  19,200,000)   100000*128
// cnt  : [19,200,000, 19,300,000)   100000 (u32)
// h1   : [19,300,000, 32,100,000)   100000*128
// zero region = first 19,300,000 words (h1 fully overwritten by gemm1)

__global__ __launch_bounds__(256) void zero_kernel(float* p, int n) {
    int i = blockIdx.x * 256 + threadIdx.x;
    if (i < n) p[i] = 0.0f;
}

// one thread handles 4 consecutive features of one edge (float4 gather)
__global__ __launch_bounds__(256) void scatter1_kernel(
    const float* __restrict__ x, const int* __restrict__ ei,
    const float* __restrict__ ew, float* __restrict__ agg1,
    unsigned* __restrict__ cnt) {
    int tid = blockIdx.x * 256 + threadIdx.x;     // E*16 threads
    int e = tid >> 4;
    int f = (tid & 15) << 2;
    int s = ei[e];
    int d = ei[NEDGES + e];
    float w = ew[e];
    float4 v = *(const float4*)(x + s * INF + f);
    float* o = agg1 + d * INF + f;
    unsafeAtomicAdd(o + 0, v.x * w);
    unsafeAtomicAdd(o + 1, v.y * w);
    unsafeAtomicAdd(o + 2, v.z * w);
    unsafeAtomicAdd(o + 3, v.w * w);
    if ((tid & 15) == 0) atomicAdd(&cnt[d], 1u);
}

__global__ __launch_bounds__(256) void scatter2_kernel(
    const float* __restrict__ h1, const int* __restrict__ ei,
    const float* __restrict__ ew, float* __restrict__ agg2) {
    int tid = blockIdx.x * 256 + threadIdx.x;     // E*32 threads
    int e = tid >> 5;
    int f = (tid & 31) << 2;
    int s = ei[e];
    int d = ei[NEDGES + e];
    float w = ew[e];
    float4 v = *(const float4*)(h1 + s * HIDF + f);
    float* o = agg2 + d * HIDF + f;
    unsafeAtomicAdd(o + 0, v.x * w);
    unsafeAtomicAdd(o + 1, v.y * w);
    unsafeAtomicAdd(o + 2, v.z * w);
    unsafeAtomicAdd(o + 3, v.w * w);
}

// in-place mean: agg[n,f] /= max(cnt[n],1)
__global__ __launch_bounds__(256) void mean_kernel(
    float* __restrict__ agg, const unsigned* __restrict__ cnt,
    int shift, int total) {
    int i = blockIdx.x * 256 + threadIdx.x;
    if (i >= total) return;
    float c = fmaxf((float)cnt[i >> shift], 1.0f);
    agg[i] = agg[i] / c;
}

// h1 = relu(mean1 @ W1l + b1 + x @ W1r)   [100000x64]@[64x128] twice, fused
// one wave per 16x16 output tile; 8 waves/block cover all 8 N-tiles
__global__ __launch_bounds__(256) void gemm1_kernel(
    const float* __restrict__ mean1, const float* __restrict__ x,
    const float* __restrict__ W1l, const float* __restrict__ b1,
    const float* __restrict__ W1r, float* __restrict__ h1) {
    int tm   = blockIdx.x;               // 0..6249
    int tn   = threadIdx.x >> 5;         // 0..7
    int lane = threadIdx.x & 31;
    int half = lane >> 4;                // lane group (0: lanes 0-15, 1: 16-31)
    int lo   = lane & 15;
    int m = tm * 16 + lo;                // A row for this lane (both halves)
    int n = tn * 16 + lo;                // B/C column for this lane

    v8f acc = {};
    for (int k = 0; k < INF; k += 4) {
        int kk = k + 2 * half;           // lanes 16-31 hold K+2..K+3
        // A fragment: 16x4 f32, one float2 (K=kk, kk+1) per lane
        v2f a0 = *(const v2f*)(mean1 + m * INF + kk);
        // B fragment: 4x16 f32, VGPR v holds K = kk+v, N = lo
        v2f bl; bl.x = W1l[kk * HIDF + n]; bl.y = W1l[(kk + 1) * HIDF + n];
        acc = __builtin_amdgcn_wmma_f32_16x16x4_f32(
            false, a0, false, bl, (short)0, acc, false, false);
        v2f a1 = *(const v2f*)(x + m * INF + kk);
        v2f br; br.x = W1r[kk * HIDF + n]; br.y = W1r[(kk + 1) * HIDF + n];
        acc = __builtin_amdgcn_wmma_f32_16x16x4_f32(
            false, a1, false, br, (short)0, acc, false, false);
    }
    float bias = b1[n];
#pragma unroll
    for (int r = 0; r < 8; ++r) {
        int row = tm * 16 + r + 8 * half;   // C VGPR r: M=r / M=8+r
        h1[row * HIDF + n] = fmaxf(acc[r] + bias, 0.0f);
    }
}

// out = mean2 @ W2l + b2 + h1 @ W2r   [100000x128]@[128x40]; N padded to 48
__global__ __launch_bounds__(96) void gemm2_kernel(
    const float* __restrict__ mean2, const float* __restrict__ h1,
    const float* __restrict__ W2l, const float* __restrict__ b2,
    const float* __restrict__ W2r, float* __restrict__ out) {
    int tm   = blockIdx.x;               // 0..6249
    int tn   = threadIdx.x >> 5;         // 0..2
    int lane = threadIdx.x & 31;
    int half = lane >> 4;
    int lo   = lane & 15;
    int m = tm * 16 + lo;
    int n = tn * 16 + lo;
    bool nv = n < NCLS;                  // column guard (loads/stores only)

    v8f acc = {};
    for (int k = 0; k < HIDF; k += 4) {
        int kk = k + 2 * half;
        v2f a0 = *(const v2f*)(mean2 + m * HIDF + kk);
        v2f bl;
        bl.x = nv ? W2l[kk * NCLS + n] : 0.0f;
        bl.y = nv ? W2l[(kk + 1) * NCLS + n] : 0.0f;
        acc = __builtin_amdgcn_wmma_f32_16x16x4_f32(
            false, a0, false, bl, (short)0, acc, false, false);
        v2f a1 = *(const v2f*)(h1 + m * HIDF + kk);
        v2f br;
        br.x = nv ? W2r[kk * NCLS + n] : 0.0f;
        br.y = nv ? W2r[(kk + 1) * NCLS + n] : 0.0f;
        acc = __builtin_amdgcn_wmma_f32_16x16x4_f32(
            false, a1, false, br, (short)0, acc, false, false);
    }
    if (nv) {
        float bias = b2[n];
#pragma unroll
        for (int r = 0; r < 8; ++r) {
            int row = tm * 16 + r + 8 * half;
            out[row * NCLS + n] = acc[r] + bias;
        }
    }
}

extern "C" void kernel_launch(void* const* d_in, const int* in_sizes, int n_in,
                              void* d_out, int out_size, void* d_ws, size_t ws_size,
                              hipStream_t stream) {
    const float* x   = (const float*)d_in[0];
    const int*   ei  = (const int*)d_in[1];
    const float* ew  = (const float*)d_in[2];
    const float* W1l = (const float*)d_in[3];
    const float* b1  = (const float*)d_in[4];
    const float* W1r = (const float*)d_in[5];
    const float* W2l = (const float*)d_in[6];
    const float* b2  = (const float*)d_in[7];
    const float* W2r = (const float*)d_in[8];
    float* out = (float*)d_out;

    float*    agg1 = (float*)d_ws;                   //  6,400,000 f32
    float*    agg2 = agg1 + (size_t)NNODES * INF;    // 12,800,000 f32
    unsigned* cnt  = (unsigned*)(agg2 + (size_t)NNODES * HIDF); // 100,000 u32
    float*    h1   = (float*)(cnt + NNODES);         // 12,800,000 f32

    const int zeroWords = NNODES * INF + NNODES * HIDF + NNODES; // 19,300,000
    zero_kernel<<<(zeroWords + 255) / 256, 256, 0, stream>>>((float*)d_ws, zeroWords);

    scatter1_kernel<<<(NEDGES * 16) / 256, 256, 0, stream>>>(x, ei, ew, agg1, cnt);
    mean_kernel<<<(NNODES * INF + 255) / 256, 256, 0, stream>>>(agg1, cnt, 6, NNODES * INF);
    gemm1_kernel<<<NNODES / 16, 256, 0, stream>>>(agg1, x, W1l, b1, W1r, h1);

    scatter2_kernel<<<(NEDGES * 32) / 256, 256, 0, stream>>>(h1, ei, ew, agg2);
    mean_kernel<<<(NNODES * HIDF + 255) / 256, 256, 0, stream>>>(agg2, cnt, 7, NNODES * HIDF);
    gemm2_kernel<<<NNODES / 16, 96, 0, stream>>>(agg2, h1, W2l, b2, W2r, out);
}